// RNN_33397665693857
// MI455X (gfx1250) — compile-verified
//
#include <hip/hip_runtime.h>
#include <cmath>

// Problem constants (from the reference)
#define NB 64
#define TT 1024
#define DD 512
#define HH 512

typedef float v2f __attribute__((ext_vector_type(2)));
typedef float v8f __attribute__((ext_vector_type(8)));
typedef unsigned int u32x4 __attribute__((ext_vector_type(4)));
typedef int i32x4 __attribute__((ext_vector_type(4)));
typedef int i32x8 __attribute__((ext_vector_type(8)));

// ---------------------------------------------------------------------------
// Kernel 1: xW[n,t,:] = x[n,t,:] @ Wx + b   (rows flattened: row = n*T + t)
// Memory-bound: 256 MB traffic vs 34 GFLOP -> ~11 us at 23.3 TB/s, while the
// math is sub-us on the WMMA pipes. So keep full fp32 precision via
// V_WMMA_F32_16X16X4_F32; no reason to quantize a bandwidth-bound GEMM.
// ---------------------------------------------------------------------------
__global__ __launch_bounds__(256) void rnn_xw_gemm(
    const float* __restrict__ x, const float* __restrict__ Wx,
    const float* __restrict__ bias, float* __restrict__ out) {
  const int lane = threadIdx.x & 31;
  const int wave = threadIdx.x >> 5;              // 0..7
  const int m    = lane & 15;                     // row/col within tile
  const int kb   = (lane >> 4) * 2;               // K sub-offset (0 or 2)

  const int row0 = blockIdx.y * 16;               // flattened (n,t) row tile
  const int col0 = blockIdx.x * 128 + wave * 16;  // H column tile

  const float* __restrict__ xrow = x + (size_t)(row0 + m) * DD;

  v8f acc = {};
#pragma unroll 4
  for (int k = 0; k < DD; k += 4) {
    v2f a, b;
    a.x = xrow[k + kb];
    a.y = xrow[k + kb + 1];
    b.x = Wx[(size_t)(k + kb) * HH + col0 + m];
    b.y = Wx[(size_t)(k + kb + 1) * HH + col0 + m];
    acc = __builtin_amdgcn_wmma_f32_16x16x4_f32(
        /*neg_a=*/false, a, /*neg_b=*/false, b,
        /*c_mod=*/(short)0, acc, /*reuse_a=*/false, /*reuse_b=*/false);
  }

  const int   col  = col0 + m;
  const float bv   = bias[col];
  const int   mb   = (lane >> 4) * 8;             // C layout: M = r + 8*(lane/16)
#pragma unroll
  for (int r = 0; r < 8; ++r) {
    const size_t row = (size_t)(row0 + mb + r);
    out[row * HH + col] = acc[r] + bv;
  }
}

// ---------------------------------------------------------------------------
// TDM helper: issue a 2-D tensor DMA (global -> LDS) per the CDNA5 D# layout.
//   group0: [1:0]=count(1), [63:32]=lds_addr, [120:64]=global_addr, [127:126]=type(2)
//   group1: [17:16]=data_size(2 -> 4B), [79:48]=tensor_dim0, [111:80]=tensor_dim1,
//           [127:112]=tile_dim0, [143:128]=tile_dim1, [207:160]=tensor_dim0_stride
// Groups 2/3 zero (2-D tensor). Tracked by TENSORcnt. This toolchain exposes
// the 6-arg builtin: (u32x4, i32x8, i32x4, i32x4, i32x8, i32 cpol).
// ---------------------------------------------------------------------------
__device__ __forceinline__ void tdm_load_2d_f32(
    unsigned lds_byte_addr, const float* gptr,
    unsigned tensor_w, unsigned tensor_h,
    unsigned tile_w, unsigned tile_h,
    unsigned long long row_stride_elems) {
  const unsigned long long ga = (unsigned long long)(size_t)gptr;
  u32x4 g0;
  g0[0] = 1u;                                        // count=1, user descriptor
  g0[1] = lds_byte_addr;                             // lds_addr
  g0[2] = (unsigned)(ga & 0xffffffffu);              // global_addr[31:0]
  g0[3] = (unsigned)((ga >> 32) & 0x01ffffffu)       // global_addr[56:32]
          | (2u << 30);                              // type = 2 ("image")
  i32x8 g1;
  g1[0] = (int)(2u << 16);                           // data_size = 4 bytes
  g1[1] = (int)((tensor_w & 0xffffu) << 16);         // tensor_dim0[15:0]
  g1[2] = (int)(((tensor_w >> 16) & 0xffffu) | ((tensor_h & 0xffffu) << 16));
  g1[3] = (int)(((tensor_h >> 16) & 0xffffu) | ((tile_w & 0xffffu) << 16));
  g1[4] = (int)(tile_h & 0xffffu);                   // tile_dim1 (tile_dim2=0)
  g1[5] = (int)(unsigned)(row_stride_elems & 0xffffffffull);  // dim0_stride lo
  g1[6] = (int)(unsigned)((row_stride_elems >> 32) & 0xffffull);
  g1[7] = 0;
  i32x4 gz4 = {0, 0, 0, 0};
  i32x8 gz8 = {0, 0, 0, 0, 0, 0, 0, 0};
  __builtin_amdgcn_tensor_load_to_lds(g0, g1, gz4, gz4, gz8, /*cpol=*/0);
}

// ---------------------------------------------------------------------------
// Kernel 2: sequential recurrence  h_t = tanh(h_{t-1} @ Wh + xW[:,t,:])
// The recurrence is batch-parallel: 4 workgroups x 16 batch rows, zero
// inter-WG sync (only a workgroup barrier per step). h tile (16x512 f32 =
// 32 KB) is LDS-resident; Wh (1 MB) streams from the 192 MB L2 every step.
// The next timestep's xW slice (16x512 f32 = 32 KB) is DMA'd into LDS by the
// Tensor Data Mover, double-buffered, overlapping the WMMA critical path.
// LDS total: 32 + 2*32 = 96 KB of the 320 KB/WGP.
// ---------------------------------------------------------------------------
__global__ __launch_bounds__(512) void rnn_recurrence(
    const float* __restrict__ h0, const float* __restrict__ Wh,
    float* __restrict__ out) {
  extern __shared__ float smem[];
  float* hbuf  = smem;                      //  0 .. 32KB : h tile (16 x 512)
  float* xwbuf[2];
  xwbuf[0] = smem + 16 * HH;                // 32 .. 64KB : xW slice, buffer 0
  xwbuf[1] = smem + 32 * HH;                // 64 .. 96KB : xW slice, buffer 1

  const int lane = threadIdx.x & 31;
  const int wave = threadIdx.x >> 5;        // 0..15
  const int m    = lane & 15;
  const int kb   = (lane >> 4) * 2;
  const int n0   = blockIdx.x * 16;         // batch-row tile
  const int col0 = wave * 32;               // this wave's column slab

  // LDS byte addresses of the two xW buffers (flat->LDS: low 32 bits).
  const unsigned ldsXw0 = (unsigned)(size_t)(void*)xwbuf[0];
  const unsigned ldsXw1 = (unsigned)(size_t)(void*)xwbuf[1];

  // Initialize LDS h tile from h0.
  for (int i = threadIdx.x; i < 16 * HH; i += 512)
    hbuf[i] = h0[(size_t)(n0 + (i >> 9)) * HH + (i & (HH - 1))];

  // Preload xW slice for t=0 into buffer 0 (rows n0..n0+15, stride T*H).
  if (wave == 0) {
    tdm_load_2d_f32(ldsXw0, out + (size_t)n0 * TT * HH,
                    HH, 16, HH, 16, (unsigned long long)TT * HH);
    __builtin_amdgcn_s_wait_tensorcnt((short)0);
  }
  __syncthreads();

  for (int t = 0; t < TT; ++t) {
    // Kick off the DMA for timestep t+1 into the other buffer while the
    // WMMA chain for step t runs. Buffer (t+1)&1 was last read at step t-1,
    // which completed before the barrier that ended step t-1.
    if (wave == 0 && t + 1 < TT) {
      tdm_load_2d_f32((t & 1) ? ldsXw0 : ldsXw1,
                      out + ((size_t)n0 * TT + (t + 1)) * HH,
                      HH, 16, HH, 16, (unsigned long long)TT * HH);
    }

    v8f acc0 = {}, acc1 = {};
    const float* __restrict__ hA = &hbuf[m * HH];
#pragma unroll 4
    for (int k = 0; k < HH; k += 4) {
      v2f a;
      a.x = hA[k + kb];
      a.y = hA[k + kb + 1];
      v2f b0, b1;
      const size_t wr0 = (size_t)(k + kb) * HH;
      const size_t wr1 = (size_t)(k + kb + 1) * HH;
      b0.x = Wh[wr0 + col0 + m];
      b0.y = Wh[wr1 + col0 + m];
      b1.x = Wh[wr0 + col0 + 16 + m];
      b1.y = Wh[wr1 + col0 + 16 + m];
      acc0 = __builtin_amdgcn_wmma_f32_16x16x4_f32(
          false, a, false, b0, (short)0, acc0, false, false);
      acc1 = __builtin_amdgcn_wmma_f32_16x16x4_f32(
          false, a, false, b1, (short)0, acc1, false, false);
    }

    __syncthreads();  // everyone done reading hbuf for this step

    // h_t = tanh(acc + xW_t) with xW_t read from the LDS slab.
    const float* __restrict__ xw = xwbuf[t & 1];
    const int mb = (lane >> 4) * 8;
#pragma unroll
    for (int r = 0; r < 8; ++r) {
      const int    row = mb + r;                               // 0..15
      const size_t g   = ((size_t)(n0 + row) * TT + t) * HH;
      const int    c0  = col0 + m;
      const int    c1  = col0 + 16 + m;
      const float v0 = tanhf(acc0[r] + xw[row * HH + c0]);
      const float v1 = tanhf(acc1[r] + xw[row * HH + c1]);
      out[g + c0] = v0;
      out[g + c1] = v1;
      hbuf[row * HH + c0] = v0;
      hbuf[row * HH + c1] = v1;
    }

    // Make sure the next slab has landed before anyone reads it.
    if (wave == 0 && t + 1 < TT) {
      __builtin_amdgcn_s_wait_tensorcnt((short)0);
    }
    __syncthreads();  // hbuf + next xw slab ready for the next step
  }
}

// ---------------------------------------------------------------------------
extern "C" void kernel_launch(void* const* d_in, const int* in_sizes, int n_in,
                              void* d_out, int out_size, void* d_ws,
                              size_t ws_size, hipStream_t stream) {
  (void)in_sizes; (void)n_in; (void)out_size; (void)d_ws; (void)ws_size;
  const float* x  = (const float*)d_in[0];
  const float* h0 = (const float*)d_in[1];
  const float* Wx = (const float*)d_in[2];
  const float* Wh = (const float*)d_in[3];
  const float* b  = (const float*)d_in[4];
  float* out = (float*)d_out;

  // Phase 1: xW + b -> d_out (memory-bound GEMM, fp32 WMMA).
  dim3 g1(HH / 128, (NB * TT) / 16);  // (4, 4096)
  rnn_xw_gemm<<<g1, 256, 0, stream>>>(x, Wx, b, out);

  // Phase 2: sequential recurrence, batch-parallel over 4 workgroups,
  // 96 KB dynamic LDS (h tile + double-buffered TDM xW slabs).
  rnn_recurrence<<<NB / 16, 512, 96 * 1024, stream>>>(h0, Wh, out);
}